// ModelNew_51677046506163
// MI455X (gfx1250) — compile-verified
//
#include <hip/hip_runtime.h>
#include <math.h>

// ConvTranspose3d(16->32, K=3, S=2, P=1) + MaxPool(6) + channel-sum, fused.
// One block per (n, pd, ph, pw). FP32 WMMA 16x16x4 on CDNA5 (wave32).

typedef float v2f __attribute__((ext_vector_type(2)));
typedef float v8f __attribute__((ext_vector_type(8)));

#define NWAVES 4
#define BLOCK (NWAVES * 32)

__global__ __launch_bounds__(BLOCK)
void convt_pool_sum(const float* __restrict__ x,     // (8,16,32,32,32)
                    const float* __restrict__ w,     // (16,32,3,3,3) IODHW
                    const float* __restrict__ bias,  // (32,)
                    float* __restrict__ out)         // (8,1,10,10,10)
{
    __shared__ float Xs[64 * 16];    // [pos 4x4x4][ci 16]   4 KB
    __shared__ float Ys[216 * 32];   // [pos 6x6x6][co 32]  27.6 KB
    __shared__ float Mred[32];

    const int b   = blockIdx.x;
    const int n   = b / 1000;
    const int rem = b % 1000;
    const int pd  = rem / 100;
    const int ph  = (rem / 10) % 10;
    const int pw  = rem % 10;
    const int tid = threadIdx.x;

    // ---- init y buffer with bias (conv-transpose adds bias before pooling) ----
    for (int e = tid; e < 216 * 32; e += BLOCK) Ys[e] = bias[e & 31];

    // ---- stage the 4x4x4 x-block (ids 3p..3p+3, all 16 ci) into LDS ----
    const float* xb = x + (size_t)n * 16 * 32768
                        + (size_t)(3 * pd) * 1024 + (3 * ph) * 32 + (3 * pw);
    for (int e = tid; e < 64 * 16; e += BLOCK) {
        int ci = e & 15, pos = e >> 4;
        int iz = pos >> 4, iy = (pos >> 2) & 3, ix = pos & 3;
        Xs[e] = xb[(size_t)ci * 32768 + iz * 1024 + iy * 32 + ix];
    }
    __syncthreads();

    const int lane  = tid & 31;
    const int wave  = tid >> 5;
    const int hl    = lane >> 4;    // which half-wave (holds K=2,3 of same rows)
    const int ln    = lane & 15;    // row (A) / col (B) index within tile
    const int kbase = hl * 2;       // K-element base inside a 4-step

    // ---- per-tap GEMMs: (27 pos x 16 ci) * (16 ci x 32 co) via v_wmma_f32_16x16x4_f32 ----
    for (int tap = wave; tap < 27; tap += NWAVES) {
        const int kd = tap / 9, kh = (tap / 3) % 3, kw = tap % 3;
        const float* wt = w + tap;  // w[ci*864 + co*27 + tap]

        for (int nt = 0; nt < 2; ++nt) {            // co tiles (2 x 16)
            // B fragments for all 4 K-steps (reused across both M tiles)
            v2f Bf[4];
            const int co = nt * 16 + ln;
#pragma unroll
            for (int s = 0; s < 4; ++s) {
                int ci0 = 4 * s + kbase;
                Bf[s].x = wt[(size_t)ci0 * 864 + co * 27];
                Bf[s].y = wt[(size_t)(ci0 + 1) * 864 + co * 27];
            }

            for (int mt = 0; mt < 2; ++mt) {        // position tiles (rows 0..26 of 32)
                const int m = mt * 16 + ln;
                v2f Af[4];
                if (m < 27) {
                    int a = m / 9, bb = (m / 3) % 3, c = m % 3;
                    int iz = a + (kd == 0), iy = bb + (kh == 0), ix = c + (kw == 0);
                    int pos = ((iz * 4) + iy) * 4 + ix;
#pragma unroll
                    for (int s = 0; s < 4; ++s) {
                        Af[s].x = Xs[pos * 16 + 4 * s + kbase];
                        Af[s].y = Xs[pos * 16 + 4 * s + kbase + 1];
                    }
                } else {
#pragma unroll
                    for (int s = 0; s < 4; ++s) { Af[s].x = 0.0f; Af[s].y = 0.0f; }
                }

                v8f C = {};
#pragma unroll
                for (int s = 0; s < 4; ++s) {
                    // 8 args: (neg_a, A, neg_b, B, c_mod, C, reuse_a, reuse_b)
                    C = __builtin_amdgcn_wmma_f32_16x16x4_f32(
                            false, Af[s], false, Bf[s], (short)0, C, false, false);
                }

                // scatter D (16x16 tile rows = tap positions, cols = co) into Ys
#pragma unroll
                for (int r = 0; r < 8; ++r) {
                    int mm = mt * 16 + r + hl * 8;   // D layout: VGPR r = rows r / r+8
                    if (mm < 27) {
                        int a = mm / 9, bb = (mm / 3) % 3, c = mm % 3;
                        int yz = 2 * (a  + (kd == 0)) + kd - 1;  // local y coord 0..5
                        int yy = 2 * (bb + (kh == 0)) + kh - 1;
                        int yx = 2 * (c  + (kw == 0)) + kw - 1;
                        int pos = (yz * 6 + yy) * 6 + yx;
                        atomicAdd(&Ys[pos * 32 + nt * 16 + ln], C[r]);  // ds_add_f32
                    }
                }
            }
        }
    }
    __syncthreads();

    // ---- 6x6x6 max-pool per channel, then channel-sum ----
    if (tid < 32) {
        float m = -INFINITY;
        for (int p = 0; p < 216; ++p) m = fmaxf(m, Ys[p * 32 + tid]);
        Mred[tid] = m;
    }
    __syncthreads();
    if (tid == 0) {
        float s = 0.0f;
        for (int c = 0; c < 32; ++c) s += Mred[c];
        out[b] = s;
    }
}

extern "C" void kernel_launch(void* const* d_in, const int* in_sizes, int n_in,
                              void* d_out, int out_size, void* d_ws, size_t ws_size,
                              hipStream_t stream) {
    const float* x    = (const float*)d_in[0];
    const float* w    = (const float*)d_in[1];
    const float* bias = (const float*)d_in[2];
    float* out        = (float*)d_out;
    convt_pool_sum<<<8000, BLOCK, 0, stream>>>(x, w, bias, out);
}